// SSM_69758858822494
// MI455X (gfx1250) — compile-verified
//
#include <hip/hip_runtime.h>
#include <math.h>

// Problem constants (match reference)
#define B_SZ   2
#define LSEQ   2048
#define DEXP   1024
#define DRANK  64
#define NST    16
#define RPROJ  (DRANK + 2*NST)   // 96
#define NCHUNK 8
#define CHUNK  (LSEQ / NCHUNK)   // 256
#define ROWS   (B_SZ * LSEQ)     // 4096

typedef __attribute__((ext_vector_type(2))) float v2f;
typedef __attribute__((ext_vector_type(8))) float v8f;

// Full-precision f32 WMMA: D(16x16) += A(16x4) * B(4x16)
__device__ __forceinline__ v8f wmma_f32_k4(v2f a, v2f b, v8f c) {
  return __builtin_amdgcn_wmma_f32_16x16x4_f32(
      /*neg_a=*/false, a, /*neg_b=*/false, b,
      /*c_mod=*/(short)0, c, /*reuse_a=*/false, /*reuse_b=*/false);
}

// ---------------------------------------------------------------------------
// Kernel P: precompute Aneg[d,n] = -exp(A_log[d,n]) and k-major transposes of
// both weight matrices so WMMA B-fragment loads are lane-coalesced:
//   xwT [k, c] = x_proj_w [c, k]   (1024 x 96)
//   dtwT[k, d] = dt_proj_w[d, k]   (64 x 1024)
// ---------------------------------------------------------------------------
__global__ void ssm_prep(const float* __restrict__ A_log, float* __restrict__ Aneg,
                         const float* __restrict__ xw,  float* __restrict__ xwT,
                         const float* __restrict__ dtw, float* __restrict__ dtwT) {
  int i = blockIdx.x * 256 + threadIdx.x;
  if (i < DEXP * NST) Aneg[i] = -__expf(A_log[i]);
  if (i < DEXP * RPROJ) {                       // coalesced writes, tiny matrix
    int k = i / RPROJ, c = i % RPROJ;
    xwT[i] = xw[(size_t)c * DEXP + k];
  }
  if (i < DRANK * DEXP) {
    int k = i / DEXP, d = i % DEXP;
    dtwT[i] = dtw[(size_t)d * DRANK + k];
  }
}

// ---------------------------------------------------------------------------
// Kernel A: fused projections with f32 WMMA.
//   One block per 16-row tile of the flattened (B*L, D) input.
//   GEMM1: proj(16x96) = x_tile(16x1024) @ x_proj_w^T + b   [waves 0..5]
//   GEMM2: dt(16x1024) = softplus(dt_r(16x64) @ dt_proj_w^T + b) [all 8 waves]
//   B (cols 64..79) and C (cols 80..95) go straight to workspace.
// ---------------------------------------------------------------------------
__global__ __launch_bounds__(256) void ssm_proj_wmma(
    const float* __restrict__ x,    const float* __restrict__ xwT,
    const float* __restrict__ xb,   const float* __restrict__ dtwT,
    const float* __restrict__ dtb,
    float* __restrict__ dtbuf, float* __restrict__ Bmbuf, float* __restrict__ Cbuf)
{
  __shared__ float sx[16 * 516];     // 16 rows x 512 k, pad 4 to break bank conflicts
  __shared__ float sproj[16 * 100];  // 16 rows x 96 cols, padded stride 100

  const int t    = threadIdx.x;
  const int wave = t >> 5;
  const int nl   = t & 15;           // lane % 16 : row M (A) / col N (B,C,D)
  const int hi   = (t >> 4) & 1;     // lane half : selects K pair (A/B frags)
  const int rowbase = blockIdx.x * 16;

  // ---- GEMM1: accumulate over K=1024 in two 512-wide LDS chunks ----
  v8f acc1;
  if (wave < 6) {
    float bv = xb[wave * 16 + nl];   // bias depends on output col only
    for (int i = 0; i < 8; ++i) acc1[i] = bv;
  }
  for (int kc = 0; kc < DEXP; kc += 512) {
    __syncthreads();                 // protect sx from previous-iter readers
    for (int i = 0; i < 8; ++i) {    // cooperative coalesced float4 fill
      int flat = t + i * 256;        // 2048 float4 = 16 rows x 128
      int r  = flat >> 7;
      int kk = (flat & 127) << 2;
      const float4 v = *(const float4*)(x + (size_t)(rowbase + r) * DEXP + kc + kk);
      float* dst = &sx[r * 516 + kk];
      dst[0] = v.x; dst[1] = v.y; dst[2] = v.z; dst[3] = v.w;
    }
    __syncthreads();
    if (wave < 6) {
      const int col = wave * 16 + nl;
      for (int k = 0; k < 512; k += 4) {
        const int kk = k + hi * 2;   // lanes 0-15: K=k,k+1 ; lanes 16-31: K=k+2,k+3
        v2f a;  a.x  = sx[nl * 516 + kk];   a.y  = sx[nl * 516 + kk + 1];
        const int gk = kc + kk;
        v2f bf;                      // k-major: 16 lanes -> 64 consecutive bytes
        bf.x = xwT[(size_t)gk * RPROJ + col];
        bf.y = xwT[(size_t)(gk + 1) * RPROJ + col];
        acc1 = wmma_f32_k4(a, bf, acc1);
      }
    }
  }
  // Scatter proj tile: VGPR i, lanes0-15 -> M=i ; lanes16-31 -> M=i+8
  if (wave < 6) {
    for (int i = 0; i < 8; ++i) {
      int M = i + hi * 8;
      sproj[M * 100 + wave * 16 + nl] = acc1[i];
      if (wave == 4) Bmbuf[(size_t)(rowbase + M) * NST + nl] = acc1[i];  // cols 64..79
      if (wave == 5) Cbuf [(size_t)(rowbase + M) * NST + nl] = acc1[i];  // cols 80..95
    }
  }
  __syncthreads();

  // ---- GEMM2: dt_r (proj cols 0..63) @ dt_proj_w^T, K=64 = 16 WMMA steps ----
  v2f a2[16];                        // hoist all A-fragments (shared across 8 col-tiles)
  for (int kt = 0; kt < 16; ++kt) {
    int kb = kt * 4 + hi * 2;
    a2[kt].x = sproj[nl * 100 + kb];
    a2[kt].y = sproj[nl * 100 + kb + 1];
  }
  for (int j = 0; j < 8; ++j) {      // each wave owns 8 col-tiles -> 1024 cols total
    const int col = wave * 128 + j * 16 + nl;
    v8f acc2;
    { float bv = dtb[col]; for (int i = 0; i < 8; ++i) acc2[i] = bv; }
    for (int kt = 0; kt < 16; ++kt) {
      int kb = kt * 4 + hi * 2;
      v2f bf;                        // k-major: coalesced across the 16 lanes
      bf.x = dtwT[(size_t)kb * DEXP + col];
      bf.y = dtwT[(size_t)(kb + 1) * DEXP + col];
      acc2 = wmma_f32_k4(a2[kt], bf, acc2);
    }
    for (int i = 0; i < 8; ++i) {    // softplus + store
      int M = i + hi * 8;
      float v  = acc2[i];
      float sp = (v > 20.0f) ? v : log1pf(__expf(v));
      dtbuf[(size_t)(rowbase + M) * DEXP + col] = sp;
    }
  }
}

// ---------------------------------------------------------------------------
// Kernel B1: per-chunk local scan (h starts at 0) -> chunk summary (S, P).
//   Block = 256 threads: 16 d-channels x 16 states; n in lane bits 0-3.
//   P = prod(exp(A*dt)) over chunk = exp(A * sum(dt))  (one exp, not 256 muls)
// ---------------------------------------------------------------------------
__global__ __launch_bounds__(256) void ssm_scan_carry(
    const float* __restrict__ x, const float* __restrict__ dtbuf,
    const float* __restrict__ Bmbuf, const float* __restrict__ Aneg,
    float* __restrict__ Sbuf, float* __restrict__ Pbuf)
{
  const int t  = threadIdx.x;
  const int bi = blockIdx.x;
  const int c  = bi & (NCHUNK - 1);
  const int dg = (bi >> 3) & 63;
  const int b  = bi >> 9;
  const int n  = t & 15;
  const int d  = dg * 16 + (t >> 4);

  const float And = Aneg[d * NST + n];
  const size_t rbase = (size_t)b * LSEQ + (size_t)c * CHUNK;
  const float* dtp = dtbuf + rbase * DEXP + d;
  const float* xp  = x     + rbase * DEXP + d;
  const float* Bp  = Bmbuf + rbase * NST + n;

  float h = 0.0f, sdt = 0.0f;
  for (int li = 0; li < CHUNK; ++li) {
    __builtin_prefetch(dtp + (size_t)(li + 8) * DEXP, 0, 1);  // global_prefetch_b8
    __builtin_prefetch(xp  + (size_t)(li + 8) * DEXP, 0, 1);  // (4KB-stride streams)
    float dtv = dtp[(size_t)li * DEXP];
    float xv  = xp [(size_t)li * DEXP];
    float Bv  = Bp [li * NST];
    float dA  = __expf(And * dtv);
    h   = fmaf(dA, h, dtv * xv * Bv);
    sdt += dtv;
  }
  const size_t si = (((size_t)b * NCHUNK + c) * DEXP + d) * NST + n;
  Sbuf[si] = h;
  Pbuf[si] = __expf(And * sdt);
}

// ---------------------------------------------------------------------------
// Kernel B2: scan the 8 chunk summaries -> incoming carry per (b,c,d,n).
// ---------------------------------------------------------------------------
__global__ void ssm_carry_combine(const float* __restrict__ Sbuf,
                                  const float* __restrict__ Pbuf,
                                  float* __restrict__ Car)
{
  const int tg = blockIdx.x * 256 + threadIdx.x;  // < B*D*N = 32768
  const int n  = tg & 15;
  const int d  = (tg >> 4) & (DEXP - 1);
  const int b  = tg >> 14;
  const size_t stride = (size_t)DEXP * NST;
  const size_t base   = (size_t)b * NCHUNK * stride + (size_t)d * NST + n;
  float cin = 0.0f;
  Car[base] = 0.0f;
  for (int c = 1; c < NCHUNK; ++c) {
    const size_t prev = base + (size_t)(c - 1) * stride;
    cin = Sbuf[prev] + Pbuf[prev] * cin;
    Car[base + (size_t)c * stride] = cin;
  }
}

// ---------------------------------------------------------------------------
// Kernel B3: rescan each chunk seeded with its carry; emit
//   out[b,l,d] = sum_n h[b,l,d,n] * C[b,l,n]  via 4 intra-wave xor shuffles.
// ---------------------------------------------------------------------------
__global__ __launch_bounds__(256) void ssm_scan_out(
    const float* __restrict__ x, const float* __restrict__ dtbuf,
    const float* __restrict__ Bmbuf, const float* __restrict__ Cbuf,
    const float* __restrict__ Aneg, const float* __restrict__ Car,
    float* __restrict__ out)
{
  const int t  = threadIdx.x;
  const int bi = blockIdx.x;
  const int c  = bi & (NCHUNK - 1);
  const int dg = (bi >> 3) & 63;
  const int b  = bi >> 9;
  const int n  = t & 15;
  const int d  = dg * 16 + (t >> 4);

  const float And = Aneg[d * NST + n];
  const size_t rbase = (size_t)b * LSEQ + (size_t)c * CHUNK;
  const float* dtp = dtbuf + rbase * DEXP + d;
  const float* xp  = x     + rbase * DEXP + d;
  const float* Bp  = Bmbuf + rbase * NST + n;
  const float* Cp  = Cbuf  + rbase * NST + n;
  float* op        = out   + rbase * DEXP + d;

  const size_t si = (((size_t)b * NCHUNK + c) * DEXP + d) * NST + n;
  float h = Car[si];

  for (int li = 0; li < CHUNK; ++li) {
    __builtin_prefetch(dtp + (size_t)(li + 8) * DEXP, 0, 1);
    __builtin_prefetch(xp  + (size_t)(li + 8) * DEXP, 0, 1);
    float dtv = dtp[(size_t)li * DEXP];
    float xv  = xp [(size_t)li * DEXP];
    float Bv  = Bp [li * NST];
    float dA  = __expf(And * dtv);
    h = fmaf(dA, h, dtv * xv * Bv);
    float y = h * Cp[li * NST];
    y += __shfl_xor(y, 1);           // reduce over n (lane bits 0..3)
    y += __shfl_xor(y, 2);
    y += __shfl_xor(y, 4);
    y += __shfl_xor(y, 8);
    if (n == 0) op[(size_t)li * DEXP] = y;
  }
}

// ---------------------------------------------------------------------------
extern "C" void kernel_launch(void* const* d_in, const int* in_sizes, int n_in,
                              void* d_out, int out_size, void* d_ws, size_t ws_size,
                              hipStream_t stream) {
  (void)in_sizes; (void)n_in; (void)out_size; (void)ws_size;
  const float* x     = (const float*)d_in[0];
  const float* xw    = (const float*)d_in[1];
  const float* xb    = (const float*)d_in[2];
  const float* dtw   = (const float*)d_in[3];
  const float* dtb   = (const float*)d_in[4];
  const float* A_log = (const float*)d_in[5];
  float* out = (float*)d_out;

  // Workspace layout (floats): ~20.3 MB total
  float* ws    = (float*)d_ws;
  float* dtbuf = ws;                                        // ROWS*DEXP      = 4M
  float* Aneg  = dtbuf + (size_t)ROWS * DEXP;               // DEXP*NST       = 16K
  float* Bm    = Aneg  + (size_t)DEXP * NST;                // ROWS*NST       = 64K
  float* Cb    = Bm    + (size_t)ROWS * NST;                // ROWS*NST       = 64K
  float* Sb    = Cb    + (size_t)ROWS * NST;                // B*NCHUNK*D*N   = 256K
  float* Pb    = Sb    + (size_t)B_SZ * NCHUNK * DEXP * NST;
  float* Car   = Pb    + (size_t)B_SZ * NCHUNK * DEXP * NST;
  float* xwT   = Car   + (size_t)B_SZ * NCHUNK * DEXP * NST; // DEXP*RPROJ   = 96K
  float* dtwT  = xwT   + (size_t)DEXP * RPROJ;               // DRANK*DEXP   = 64K

  hipLaunchKernelGGL(ssm_prep, dim3((DEXP * RPROJ + 255) / 256), dim3(256), 0, stream,
                     A_log, Aneg, xw, xwT, dtw, dtwT);
  hipLaunchKernelGGL(ssm_proj_wmma, dim3(ROWS / 16), dim3(256), 0, stream,
                     x, xwT, xb, dtwT, dtb, dtbuf, Bm, Cb);
  hipLaunchKernelGGL(ssm_scan_carry, dim3(B_SZ * (DEXP / 16) * NCHUNK), dim3(256), 0, stream,
                     x, dtbuf, Bm, Aneg, Sb, Pb);
  hipLaunchKernelGGL(ssm_carry_combine, dim3((B_SZ * DEXP * NST) / 256), dim3(256), 0, stream,
                     Sb, Pb, Car);
  hipLaunchKernelGGL(ssm_scan_out, dim3(B_SZ * (DEXP / 16) * NCHUNK), dim3(256), 0, stream,
                     x, dtbuf, Bm, Cb, Aneg, Car, out);
}